// RandomConv_23519240913600
// MI455X (gfx1250) — compile-verified
//
#include <hip/hip_runtime.h>

typedef __attribute__((ext_vector_type(2))) float v2f;
typedef __attribute__((ext_vector_type(8))) float v8f;

#define NF        3
#define CIN       9              // NF * IMG_C
#define H         128
#define W         128
#define TILE_H    8
#define XS_ROWS   (TILE_H + 2)   // 10
#define XS_COLS   (W + 2)        // 130
#define XS_CH     (XS_ROWS * XS_COLS)  // 1300
#define XS_TOTAL  (CIN * XS_CH)        // 11700
#define KDIM      81             // 9 in-ch * 3x3 im2col depth
#define KPAD      84             // pad to multiple of 4 for 16x16x4 WMMA
#define AROWS     16
#define NKSTEP    (KPAD / 4)     // 21

// Compile-time im2col LDS offset for K-index q (relative to row*XS_COLS + col)
__device__ constexpr int imoff(int q) {
    return (q / 9) * XS_CH + ((q % 9) / 3) * XS_COLS + (q % 3);
}
__device__ constexpr int imoff_c(int q) {   // clamped so speculative loads stay in Xs
    return imoff(q < KDIM ? q : KDIM - 1);
}

__global__ __launch_bounds__(256)
void randconv_wmma_f32(const float* __restrict__ x,
                       const float* __restrict__ wgt,
                       float* __restrict__ out) {
    __shared__ float Xs[XS_TOTAL];        // staged replicate-padded input tile
    __shared__ float Aw[AROWS * KPAD];    // block-diagonal weight matrix (x 1/255)

    const int n   = blockIdx.y;
    const int h0  = blockIdx.x * TILE_H;
    const int tid = threadIdx.x;

    // ---- phase 1a: zero A (scatter below only fills 243 of 1344 entries)
    for (int i = tid; i < AROWS * KPAD; i += 256) Aw[i] = 0.0f;

    // ---- phase 1b: stage input tile via async global->LDS DMA (ASYNCcnt path).
    // Replicate padding = clamped per-lane global addresses; /255 is folded into weights,
    // so this is a pure copy: global_load_async_to_lds_b32 moves each element straight
    // into LDS without touching data VGPRs.
    {
        const float* xb = x + (size_t)n * CIN * H * W;
        int cg = 0, rr = 0, cc = tid;
        while (cc >= XS_COLS) { cc -= XS_COLS; ++rr; }   // tid < 256 -> at most 1 iter
        for (int i = tid; i < XS_TOTAL; i += 256) {
            int gh = h0 - 1 + rr; gh = gh < 0 ? 0 : (gh > H - 1 ? H - 1 : gh);
            int gw = cc - 1;      gw = gw < 0 ? 0 : (gw > W - 1 ? W - 1 : gw);
            const float* gp = xb + cg * (H * W) + gh * W + gw;
            unsigned lds = (unsigned)(uintptr_t)(&Xs[i]);
            asm volatile("global_load_async_to_lds_b32 %0, %1, off"
                         :: "v"(lds), "v"(gp) : "memory");
            // advance flat index by 256 = 1*130 + 126 with carry propagation
            cc += 126; rr += 1;
            if (cc >= XS_COLS) { cc -= XS_COLS; rr += 1; }
            if (rr >= XS_ROWS) { rr -= XS_ROWS; cg += 1; }
        }
        asm volatile("s_wait_asynccnt 0x0" ::: "memory");
    }
    __syncthreads();

    // ---- phase 2: scatter per-sample weights into A.
    // out channel j = t*3+o reads in channels t*3+c; same 3x3x3x3 weights for every frame t.
    if (tid < NF * KDIM) {
        int t  = tid / KDIM;
        int r  = tid - t * KDIM;          // r = o*27 + c*9 + kh*3 + kw
        int o  = r / 27;
        int c  = (r / 9) % 3;
        int kh = (r / 3) % 3;
        int kw = r % 3;
        float wv = wgt[(size_t)n * KDIM + r] * (1.0f / 255.0f);
        int row = t * 3 + o;
        int col = (t * 3 + c) * 9 + kh * 3 + kw;
        Aw[row * KPAD + col] = wv;
    }
    __syncthreads();

    // ---- phase 3: WMMA. A rows = out channels, B cols = pixels (so D stores coalesce).
    const int wv_id = tid >> 5;           // wave32
    const int lane  = tid & 31;
    const int m     = lane & 15;          // A row / B col(pixel) / D col
    const int khalf = lane >> 4;          // K pair {2*khalf, 2*khalf+1} within each k-step

    // A fragments are identical for every pixel group this wave processes -> load once.
    v2f afrag[NKSTEP];
#pragma unroll
    for (int k = 0; k < NKSTEP; ++k)
        afrag[k] = *(const v2f*)(&Aw[m * KPAD + 4 * k + 2 * khalf]);  // 8B aligned

    for (int g = wv_id; g < TILE_H * (W / 16); g += 8) {
        const int row = g >> 3;           // tile row 0..7
        const int cb  = g & 7;            // 16-pixel column block
        const int bbase = row * XS_COLS + cb * 16 + m;
        v8f acc = {0.f, 0.f, 0.f, 0.f, 0.f, 0.f, 0.f, 0.f};

#pragma unroll
        for (int k = 0; k < NKSTEP; ++k) {
            // B fragment: compile-time im2col offsets, khalf-selected (1 cndmask each)
            const int q0   = 4 * k + 2 * khalf;
            const int off0 = khalf ? imoff_c(4 * k + 2) : imoff_c(4 * k + 0);
            const int off1 = khalf ? imoff_c(4 * k + 3) : imoff_c(4 * k + 1);
            float b0 = (q0     < KDIM) ? Xs[bbase + off0] : 0.f;
            float b1 = (q0 + 1 < KDIM) ? Xs[bbase + off1] : 0.f;  // compile-time 0 for k=20
            v2f b = {b0, b1};
            acc = __builtin_amdgcn_wmma_f32_16x16x4_f32(false, afrag[k], false, b,
                                                        (short)0, acc, false, false);
        }

        // D layout: VGPR r, lane half kh -> channel (r + 8*khalf), pixel = lane&15.
        // Lanes within a half write 16 contiguous floats -> coalesced 64B stores.
        float* ob = out + (size_t)n * CIN * H * W + (size_t)(h0 + row) * W + cb * 16 + m;
#pragma unroll
        for (int r8 = 0; r8 < 8; ++r8) {
            int ch = r8 + 8 * khalf;
            if (ch < CIN) {
                float v = acc[r8];
                float s = 255.0f * __builtin_amdgcn_rcpf(1.0f + __expf(-v)); // sigmoid*255
                ob[(size_t)ch * (H * W)] = s;
            }
        }
    }
}

extern "C" void kernel_launch(void* const* d_in, const int* in_sizes, int n_in,
                              void* d_out, int out_size, void* d_ws, size_t ws_size,
                              hipStream_t stream) {
    const float* x   = (const float*)d_in[0];   // [n, 9, 128, 128] f32
    const float* wgt = (const float*)d_in[1];   // [n, 3, 3, 3, 3]  f32
    float* out = (float*)d_out;                 // [n, 9, 128, 128] f32
    int nimg = in_sizes[0] / (CIN * H * W);     // 512
    dim3 grid(H / TILE_H, nimg);
    randconv_wmma_f32<<<grid, 256, 0, stream>>>(x, wgt, out);
}